// S4Layer_38448547234651
// MI455X (gfx1250) — compile-verified
//
#include <hip/hip_runtime.h>

// ---------------------------------------------------------------------------
// S4 layer for MI455X (gfx1250, wave32).
//  - setup: GJ inverse + Abar^(2^19) via V_WMMA_F32_16X16X4_F32 (f32 matrix pipe)
//  - cauchy: VALU-bound kernel, one thread per frequency
//  - FFTs: four-step; 8 columns per workgroup in LDS, gathered with
//    GLOBAL_LOAD_ASYNC_TO_LDS_B64 (ASYNCcnt) so strided global reads become
//    contiguous 64B segments and never bounce through VGPRs.
// ---------------------------------------------------------------------------

#define LSEQ   (1 << 19)   // sequence length
#define L2SEQ  (1 << 20)   // padded conv length
#define NS     64          // state size
#define MSTR   65          // padded LDS row stride (float2) for the matmul tiles
#define CW     8           // FFT columns per workgroup

typedef float v2f __attribute__((ext_vector_type(2)));
typedef float v8f __attribute__((ext_vector_type(8)));

__device__ __forceinline__ float2 cmulc(float2 a, float2 b) {
  return make_float2(a.x * b.x - a.y * b.y, a.x * b.y + a.y * b.x);
}
__device__ __forceinline__ float2 cdivc(float2 a, float2 b) {
  float inv = 1.0f / (b.x * b.x + b.y * b.y);
  return make_float2((a.x * b.x + a.y * b.y) * inv, (a.y * b.x - a.x * b.y) * inv);
}

// ---------------------------------------------------------------------------
// Complex 64x64 matmul C = A*B, all operands in LDS (row stride MSTR float2).
// 8 waves, 16 output tiles of 16x16, V_WMMA_F32_16X16X4_F32.
// ---------------------------------------------------------------------------
__device__ __forceinline__ void cmm64(const float2* Am, const float2* Bm, float2* Cm) {
  const int lane  = threadIdx.x & 31;
  const int wave  = threadIdx.x >> 5;
  const int mloc  = lane & 15;
  const int khalf = (lane >> 4) << 1;   // 0 or 2

  for (int t = wave; t < 16; t += 8) {
    const int tr = (t >> 2) << 4;   // tile row base
    const int tc = (t & 3) << 4;    // tile col base
    v8f cr = {0.f, 0.f, 0.f, 0.f, 0.f, 0.f, 0.f, 0.f};
    v8f ci = {0.f, 0.f, 0.f, 0.f, 0.f, 0.f, 0.f, 0.f};
    for (int kb = 0; kb < 16; ++kb) {
      const int k0 = (kb << 2) + khalf;
      const float2 a0 = Am[(tr + mloc) * MSTR + k0];
      const float2 a1 = Am[(tr + mloc) * MSTR + k0 + 1];
      const float2 b0 = Bm[k0 * MSTR + tc + mloc];
      const float2 b1 = Bm[(k0 + 1) * MSTR + tc + mloc];
      v2f aR  = {a0.x, a1.x};
      v2f aI  = {a0.y, a1.y};
      v2f aIn = {-a0.y, -a1.y};
      v2f bR  = {b0.x, b1.x};
      v2f bI  = {b0.y, b1.y};
      cr = __builtin_amdgcn_wmma_f32_16x16x4_f32(false, aR,  false, bR, (short)0, cr, false, false);
      cr = __builtin_amdgcn_wmma_f32_16x16x4_f32(false, aIn, false, bI, (short)0, cr, false, false);
      ci = __builtin_amdgcn_wmma_f32_16x16x4_f32(false, aR,  false, bI, (short)0, ci, false, false);
      ci = __builtin_amdgcn_wmma_f32_16x16x4_f32(false, aI,  false, bR, (short)0, ci, false, false);
    }
    const int nc = tc + (lane & 15);
    const int mb = tr + ((lane >> 4) << 3);
#pragma unroll
    for (int g = 0; g < 8; ++g)
      Cm[(mb + g) * MSTR + nc] = make_float2(cr[g], ci[g]);
  }
  __syncthreads();
}

// ---------------------------------------------------------------------------
// Setup kernel: one workgroup, 256 threads, dynamic LDS.
// ---------------------------------------------------------------------------
extern "C" __global__ void __launch_bounds__(256)
s4_setup_kernel(const float2* __restrict__ A, const float2* __restrict__ p,
                const float2* __restrict__ q, const float* __restrict__ Bv,
                const float* __restrict__ Cv, float2* __restrict__ nums_out) {
  extern __shared__ float2 sm[];
  float2* aug = sm;             // 64*128 = 8192 f2
  float2* fac = sm + 8192;      // 64 f2
  float2* mA  = sm;             // 64*MSTR
  float2* mB  = sm + 64 * MSTR;
  float2* mC  = sm + 2 * 64 * MSTR;

  const int tid = threadIdx.x;
  const float hh = 0.5f / (float)LSEQ;   // step/2

  for (int idx = tid; idx < NS * NS; idx += 256) {
    const int r = idx >> 6, c = idx & 63;
    const float2 a = A[idx];
    const float d = (r == c) ? 1.f : 0.f;
    aug[r * 128 + c]      = make_float2(d - hh * a.x, -hh * a.y);
    aug[r * 128 + 64 + c] = make_float2(d, 0.f);
  }
  __syncthreads();

  // Gauss-Jordan (no pivoting: I - (tiny)*A is strongly diag-dominant)
  for (int col = 0; col < NS; ++col) {
    const float2 piv = aug[col * 128 + col];
    const float id = 1.f / (piv.x * piv.x + piv.y * piv.y);
    const float2 pinv = make_float2(piv.x * id, -piv.y * id);
    for (int j = tid; j < 128; j += 256)
      aug[col * 128 + j] = cmulc(aug[col * 128 + j], pinv);
    __syncthreads();
    if (tid < 64) fac[tid] = aug[tid * 128 + col];
    __syncthreads();
    for (int idx = tid; idx < NS * 128; idx += 256) {
      const int r = idx >> 7, j = idx & 127;
      if (r != col) {
        const float2 f = fac[r];
        const float2 pr = aug[col * 128 + j];
        const float2 v = aug[idx];
        aug[idx] = make_float2(v.x - (f.x * pr.x - f.y * pr.y),
                               v.y - (f.x * pr.y + f.y * pr.x));
      }
    }
    __syncthreads();
  }

  float2 regs[16];
#pragma unroll
  for (int t = 0; t < 16; ++t) {
    const int idx = tid + t * 256;
    regs[t] = aug[(idx >> 6) * 128 + 64 + (idx & 63)];
  }
  __syncthreads();
#pragma unroll
  for (int t = 0; t < 16; ++t) {
    const int idx = tid + t * 256;
    mB[(idx >> 6) * MSTR + (idx & 63)] = regs[t];
  }
  for (int idx = tid; idx < NS * NS; idx += 256) {
    const int r = idx >> 6, c = idx & 63;
    const float2 a = A[idx];
    const float d = (r == c) ? 1.f : 0.f;
    mA[r * MSTR + c] = make_float2(d + hh * a.x, hh * a.y);
  }
  __syncthreads();

  cmm64(mB, mA, mC);                       // Abar -> mC
  for (int idx = tid; idx < NS * NS; idx += 256)
    mA[(idx >> 6) * MSTR + (idx & 63)] = mC[(idx >> 6) * MSTR + (idx & 63)];
  __syncthreads();

  float2* src = mA; float2* dst = mB;
  for (int it = 0; it < 19; ++it) {        // Abar^(2^19)
    cmm64(src, src, dst);
    float2* tmp = src; src = dst; dst = tmp;
  }

  if (tid < NS) {
    const int n = tid;
    float re = 0.f, im = 0.f;
    for (int m = 0; m < NS; ++m) {
      const float2 pv = src[m * MSTR + n];
      const float d = (m == n) ? 1.f : 0.f;
      const float cm = Cv[m];
      re += cm * (d - pv.x);
      im += cm * pv.y;
    }
    const float2 a0 = make_float2(re, -im);         // conj(Ct)
    const float2 qv = q[n];
    const float2 a1 = make_float2(qv.x, -qv.y);     // conj(q)
    const float2 b0 = make_float2(Bv[n], 0.f);
    const float2 b1 = p[n];
    nums_out[0 * NS + n] = cmulc(a0, b0);
    nums_out[1 * NS + n] = cmulc(a0, b1);
    nums_out[2 * NS + n] = cmulc(a1, b0);
    nums_out[3 * NS + n] = cmulc(a1, b1);
  }
}

// ---------------------------------------------------------------------------
// Cauchy kernel: one thread per frequency l. VALU-bound; Gamma/nums in LDS.
// ---------------------------------------------------------------------------
extern "C" __global__ void __launch_bounds__(256)
s4_cauchy_kernel(const float2* __restrict__ Gamma, const float2* __restrict__ nums,
                 float2* __restrict__ atR) {
  __shared__ float2 sg[NS];
  __shared__ float2 sn[4 * NS];
  const int tid = threadIdx.x;
  if (tid < NS) sg[tid] = Gamma[tid];
  sn[tid] = nums[tid];
  __syncthreads();

  const int l = blockIdx.x * 256 + tid;
  const float theta = 6.28318530717958647692f * ((float)l / (float)LSEQ);
  float sr, cr; __sincosf(theta, &sr, &cr);
  const float2 onePr = make_float2(1.f + cr, sr);
  const float2 oneMr = make_float2(1.f - cr, -sr);
  float2 f = cdivc(oneMr, onePr);
  const float twoL = 2.f * (float)LSEQ;   // 2/step
  f.x *= twoL; f.y *= twoL;

  float2 k00 = make_float2(0.f, 0.f), k01 = k00, k10 = k00, k11 = k00;
#pragma unroll 4
  for (int n = 0; n < NS; ++n) {
    const float2 g = sg[n];
    const float2 d = make_float2(f.x - g.x, f.y - g.y);
    const float inv = 1.f / (d.x * d.x + d.y * d.y);
    const float2 rc = make_float2(d.x * inv, -d.y * inv);
    k00 = make_float2(k00.x + sn[n].x * rc.x - sn[n].y * rc.y,
                      k00.y + sn[n].x * rc.y + sn[n].y * rc.x);
    k01 = make_float2(k01.x + sn[NS + n].x * rc.x - sn[NS + n].y * rc.y,
                      k01.y + sn[NS + n].x * rc.y + sn[NS + n].y * rc.x);
    k10 = make_float2(k10.x + sn[2 * NS + n].x * rc.x - sn[2 * NS + n].y * rc.y,
                      k10.y + sn[2 * NS + n].x * rc.y + sn[2 * NS + n].y * rc.x);
    k11 = make_float2(k11.x + sn[3 * NS + n].x * rc.x - sn[3 * NS + n].y * rc.y,
                      k11.y + sn[3 * NS + n].x * rc.y + sn[3 * NS + n].y * rc.x);
  }
  const float2 onePk11 = make_float2(1.f + k11.x, k11.y);
  const float2 t = cdivc(cmulc(k01, k10), onePk11);
  const float2 inner = make_float2(k00.x - t.x, k00.y - t.y);
  atR[l] = cmulc(cdivc(make_float2(2.f, 0.f), onePr), inner);
}

// ---------------------------------------------------------------------------
// Four-step FFT, N = N1*N2, 8 columns per workgroup.
// Pass 1 (block = 8 consecutive n1): async-gather tile[8][N2] from
// in[n1 + N1*n2] (8 lanes -> 64B contiguous), FFT each column over n2,
// twiddle by W_N^(n1*k2), store coalesced to out[n1*N2 + k2].
// Pass 2 (block = 8 consecutive k2): async-gather tile[8][N1] from
// in[n1*N2 + k2], FFT over n1, store out[k1*N2 + k2] (64B segments).
// Column stride padded (+4 float2) so the 8 columns hit distinct LDS banks.
// ---------------------------------------------------------------------------
__device__ __forceinline__ void async_gather_f2(unsigned lds_off, const float2* gp) {
  const unsigned long long ga = (unsigned long long)gp;
  asm volatile("global_load_async_to_lds_b64 %0, %1, off"
               :: "v"(lds_off), "v"(ga) : "memory");
}
__device__ __forceinline__ void async_wait_all() {
  asm volatile("s_wait_asynccnt 0x0" ::: "memory");
}

__device__ __forceinline__ void fft_cols(float2* tile, int tstr, int n, int logn, float sgn) {
  const int tid = threadIdx.x;
  // bit-reverse each column
  for (int idx = tid; idx < CW * n; idx += 256) {
    const int c = idx & (CW - 1);
    const int r = idx >> 3;
    const int j = (int)(__brev((unsigned)r) >> (32 - logn));
    if (r < j) {
      float2 t0 = tile[c * tstr + r];
      tile[c * tstr + r] = tile[c * tstr + j];
      tile[c * tstr + j] = t0;
    }
  }
  __syncthreads();
  for (int len = 2; len <= n; len <<= 1) {
    const unsigned half = (unsigned)(len >> 1);
    const float ang0 = sgn * 6.28318530717958647692f / (float)len;
    for (int b = tid; b < CW * (n >> 1); b += 256) {
      const int c = b & (CW - 1);
      const unsigned bb = (unsigned)(b >> 3);
      const unsigned grp = bb / half;
      const unsigned j = bb - grp * half;
      const int i0 = c * tstr + (int)(grp * (unsigned)len + j);
      const int i1 = i0 + (int)half;
      float sw, cwv; __sincosf(ang0 * (float)j, &sw, &cwv);
      const float2 u = tile[i0], v = tile[i1];
      const float2 w = make_float2(v.x * cwv - v.y * sw, v.x * sw + v.y * cwv);
      tile[i0] = make_float2(u.x + w.x, u.y + w.y);
      tile[i1] = make_float2(u.x - w.x, u.y - w.y);
    }
    __syncthreads();
  }
}

extern "C" __global__ void __launch_bounds__(256)
s4_fft_pass1(const float2* __restrict__ in, float2* __restrict__ out,
             int N1, int N2, int logN2, float sgn) {
  extern __shared__ float2 tile[];
  const int tstr = N2 + 4;
  const int n1base = blockIdx.x * CW;
  const int tid = threadIdx.x;
  const unsigned ldsbase = __builtin_amdgcn_groupstaticsize();

  for (int idx = tid; idx < CW * N2; idx += 256) {
    const int c = idx & (CW - 1);
    const int n2 = idx >> 3;
    const unsigned loff = ldsbase + (unsigned)(c * tstr + n2) * 8u;
    async_gather_f2(loff, in + (size_t)(n1base + c) + (size_t)N1 * (size_t)n2);
  }
  async_wait_all();
  __syncthreads();

  fft_cols(tile, tstr, N2, logN2, sgn);

  const long long Nll = (long long)N1 * (long long)N2;
  const float invN = 1.0f / (float)Nll;
  for (int idx = tid; idx < CW * N2; idx += 256) {
    const int c = idx >> logN2;             // k2 fast -> coalesced stores
    const int k2 = idx & (N2 - 1);
    const int n1 = n1base + c;
    const long long prod = ((long long)n1 * (long long)k2) % Nll;
    const float ang = sgn * 6.28318530717958647692f * ((float)prod * invN);
    float sw, cwv; __sincosf(ang, &sw, &cwv);
    const float2 v = tile[c * tstr + k2];
    out[(size_t)n1 * (size_t)N2 + (size_t)k2] =
        make_float2(v.x * cwv - v.y * sw, v.x * sw + v.y * cwv);
  }
}

extern "C" __global__ void __launch_bounds__(256)
s4_fft_pass2(const float2* __restrict__ in, float2* __restrict__ out,
             int N1, int N2, int logN1, float sgn) {
  extern __shared__ float2 tile[];
  const int tstr = N1 + 4;
  const int k2base = blockIdx.x * CW;
  const int tid = threadIdx.x;
  const unsigned ldsbase = __builtin_amdgcn_groupstaticsize();

  for (int idx = tid; idx < CW * N1; idx += 256) {
    const int c = idx & (CW - 1);
    const int n1 = idx >> 3;
    const unsigned loff = ldsbase + (unsigned)(c * tstr + n1) * 8u;
    async_gather_f2(loff, in + (size_t)n1 * (size_t)N2 + (size_t)(k2base + c));
  }
  async_wait_all();
  __syncthreads();

  fft_cols(tile, tstr, N1, logN1, sgn);

  for (int idx = tid; idx < CW * N1; idx += 256) {
    const int c = idx & (CW - 1);           // c fast -> 64B store segments
    const int k1 = idx >> 3;
    out[(size_t)k1 * (size_t)N2 + (size_t)(k2base + c)] = tile[c * tstr + k1];
  }
}

// K[j] = real(fft(atRoots)[j]) / L  (reversal identity), zero-padded to 2L.
extern "C" __global__ void __launch_bounds__(256)
s4_make_kc(const float2* __restrict__ Fa, float2* __restrict__ Kc) {
  const int i = blockIdx.x * 256 + threadIdx.x;
  const float invL = 1.0f / (float)LSEQ;
  Kc[i] = (i < LSEQ) ? make_float2(Fa[i].x * invL, 0.f) : make_float2(0.f, 0.f);
}

extern "C" __global__ void __launch_bounds__(256)
s4_pack_y(const float* __restrict__ y, float2* __restrict__ yc) {
  const int i = blockIdx.x * 256 + threadIdx.x;
  yc[i] = (i < LSEQ) ? make_float2(y[i], 0.f) : make_float2(0.f, 0.f);
}

extern "C" __global__ void __launch_bounds__(256)
s4_cmul(float2* __restrict__ a, const float2* __restrict__ b) {
  const int i = blockIdx.x * 256 + threadIdx.x;
  a[i] = cmulc(a[i], b[i]);
}

extern "C" __global__ void __launch_bounds__(256)
s4_final(const float2* __restrict__ Pi, const float* __restrict__ y,
         const float* __restrict__ D, float* __restrict__ out) {
  const int i = blockIdx.x * 256 + threadIdx.x;
  out[i] = Pi[i].x * (1.0f / (float)L2SEQ) + D[0] * y[i];
}

// ---------------------------------------------------------------------------
extern "C" void kernel_launch(void* const* d_in, const int* in_sizes, int n_in,
                              void* d_out, int out_size, void* d_ws, size_t ws_size,
                              hipStream_t stream) {
  (void)in_sizes; (void)n_in; (void)out_size; (void)ws_size;
  const float2* A     = (const float2*)d_in[0];
  const float2* p     = (const float2*)d_in[1];
  const float2* q     = (const float2*)d_in[2];
  const float2* Gamma = (const float2*)d_in[3];
  const float*  Bv    = (const float*)d_in[4];
  const float*  Cv    = (const float*)d_in[5];
  const float*  Dv    = (const float*)d_in[6];
  const float*  y     = (const float*)d_in[7];
  float* out = (float*)d_out;

  float2* W    = (float2*)d_ws;
  float2* nums = W;                                 // 256 (pad to 1024)
  float2* atR  = W + 1024;                          // 2^19
  float2* B1   = W + 1024 + LSEQ;                   // 2^20 each
  float2* B2   = B1 + L2SEQ;
  float2* B3   = B2 + L2SEQ;
  float2* B4   = B3 + L2SEQ;

  const size_t ldsSetup  = (size_t)(3 * 64 * MSTR) * sizeof(float2); // ~97.5 KB
  const size_t lds1024   = (size_t)(CW * (1024 + 4)) * sizeof(float2); // 65792 B
  const size_t lds512    = (size_t)(CW * (512 + 4)) * sizeof(float2);  // 33024 B

  s4_setup_kernel<<<1, 256, ldsSetup, stream>>>(A, p, q, Bv, Cv, nums);
  s4_cauchy_kernel<<<LSEQ / 256, 256, 0, stream>>>(Gamma, nums, atR);

  // K = real(fft_{2^19}(atRoots))/L, zero-padded to 2^20  (N1=512, N2=1024)
  s4_fft_pass1<<<512 / CW, 256, lds1024, stream>>>(atR, B1, 512, 1024, 10, -1.f);
  s4_fft_pass2<<<1024 / CW, 256, lds512, stream>>>(B1, B2, 512, 1024, 9, -1.f);
  s4_make_kc<<<L2SEQ / 256, 256, 0, stream>>>(B2, B3);
  s4_pack_y<<<L2SEQ / 256, 256, 0, stream>>>(y, B1);

  // Yf = fft_{2^20}(y_pad)
  s4_fft_pass1<<<1024 / CW, 256, lds1024, stream>>>(B1, B2, 1024, 1024, 10, -1.f);
  s4_fft_pass2<<<1024 / CW, 256, lds1024, stream>>>(B2, B4, 1024, 1024, 10, -1.f);
  // Kf = fft_{2^20}(K_pad)
  s4_fft_pass1<<<1024 / CW, 256, lds1024, stream>>>(B3, B2, 1024, 1024, 10, -1.f);
  s4_fft_pass2<<<1024 / CW, 256, lds1024, stream>>>(B2, B1, 1024, 1024, 10, -1.f);
  // P = Yf .* Kf
  s4_cmul<<<L2SEQ / 256, 256, 0, stream>>>(B4, B1);
  // conv = ifft(P).real (scaled in final)
  s4_fft_pass1<<<1024 / CW, 256, lds1024, stream>>>(B4, B2, 1024, 1024, 10, +1.f);
  s4_fft_pass2<<<1024 / CW, 256, lds1024, stream>>>(B2, B3, 1024, 1024, 10, +1.f);

  s4_final<<<LSEQ / 256, 256, 0, stream>>>(B3, y, Dv, out);
}